// GraphSageConvolution_38371237822943
// MI455X (gfx1250) — compile-verified
//
#include <hip/hip_runtime.h>
#include <math.h>

#define N_SRC   200000
#define N_DST   50000
#define N_EDGES 800000
#define N_IN    256
#define N_OUT   128

typedef __attribute__((ext_vector_type(2))) float v2f;
typedef __attribute__((ext_vector_type(8))) float v8f;

// ---------------------------------------------------------------------------
// Kernel 0: zero the feat scratch buffer (50000 x 256 f32) with float4 stores.
// 3,200,000 float4 = 12500 blocks x 256 threads, exact cover.
// ---------------------------------------------------------------------------
__global__ void sage_zero_feat(float4* __restrict__ feat4) {
    int i = blockIdx.x * blockDim.x + threadIdx.x;
    feat4[i] = make_float4(0.f, 0.f, 0.f, 0.f);
}

// ---------------------------------------------------------------------------
// Kernel 0b: pack both weight matrices into column-major WT[128][256] so a
// WMMA B-fragment (W[kk][n], W[kk+1][n]) becomes one contiguous float2 load.
// 2 * 256*128 elements = 65536 threads = 256 blocks x 256.
// ---------------------------------------------------------------------------
__global__ void sage_pack_w(const float* __restrict__ W_b,
                            const float* __restrict__ W_w,
                            float* __restrict__ WT_b,
                            float* __restrict__ WT_w) {
    int id = blockIdx.x * 256 + threadIdx.x;     // 0..65535
    int m  = id >> 15;                           // 0: W_b, 1: W_w
    int r  = id & 32767;
    int k  = r >> 7;                             // 0..255
    int n  = r & 127;                            // 0..127
    if (m == 0) WT_b[n * N_IN + k] = W_b[k * N_OUT + n];
    else        WT_w[n * N_IN + k] = W_w[k * N_OUT + n];
}

// ---------------------------------------------------------------------------
// Kernel 1: SpMM scatter. 64 threads per edge, each thread handles 4 features
// via a coalesced float4 gather of the source row and 4 f32 atomics into feat
// (L2-resident, 51 MB). 800000 edges * 64 threads / 256 = 200000 blocks exact.
// ---------------------------------------------------------------------------
__global__ void sage_spmm(const float*  __restrict__ x,
                          const int*    __restrict__ erows,
                          const int*    __restrict__ ecols,
                          const float*  __restrict__ evals,
                          float*        __restrict__ feat) {
    int gid  = blockIdx.x * 256 + threadIdx.x;
    int e    = gid >> 6;          // edge index
    int l64  = gid & 63;          // 64 lanes per edge, 4 floats each
    int r    = erows[e];
    int c    = ecols[e];
    float v  = evals[e];
    float4 xv = *(const float4*)(x + (size_t)c * N_IN + l64 * 4);
    float* dst = feat + (size_t)r * N_IN + l64 * 4;
    atomicAdd(dst + 0, v * xv.x);
    atomicAdd(dst + 1, v * xv.y);
    atomicAdd(dst + 2, v * xv.z);
    atomicAdd(dst + 3, v * xv.w);
}

// ---------------------------------------------------------------------------
// Kernel 2: fused dual GEMM (WMMA f32 16x16x4) + ELU + row layernorm + affine.
// One 256-thread block (8 waves) owns 16 destination rows.
//   Stage A: feat tile staged with CDNA5 async global->LDS b128 copies
//            (ASYNCcnt); gathered x[sampled] rows staged synchronously.
//            LDS stride 260 words kills bank conflicts on A-fragment reads.
//   Stage B: wave w computes out cols [16w,16w+16) of the self half (x@W_b)
//            and of the neigh half (feat@W_w); K=256 in 64 chained WMMAs,
//            accumulator seeded with the bias, B fragments as float2 from WT.
//   Stage C: wave w normalizes rows 2w,2w+1 across the 256-wide concat row.
// ---------------------------------------------------------------------------
#define XS_STRIDE 260   // 256 + 4: 4*row bank rotation, keeps 16B alignment

__global__ __launch_bounds__(256)
void sage_fused(const float* __restrict__ x,
                const int*   __restrict__ sampled,
                const float* __restrict__ feat,
                const float* __restrict__ WT_w,
                const float* __restrict__ b_w,
                const float* __restrict__ WT_b,
                const float* __restrict__ b_b,
                const float* __restrict__ scale,
                const float* __restrict__ offset,
                float*       __restrict__ out) {
    __shared__ float xs[16 * XS_STRIDE];   // gathered x rows
    __shared__ float fs[16 * XS_STRIDE];   // feat rows
    __shared__ float os[16 * 256];         // concat output tile (self|neigh)

    const int t      = threadIdx.x;
    const int m_base = blockIdx.x * 16;

    // ---- Stage A: stage 16x256 tiles into LDS ----------------------------
    #pragma unroll
    for (int i = 0; i < 4; ++i) {
        int idx = t + 256 * i;          // 0..1023 over 16 rows x 64 float4
        int row = idx >> 6;
        int c4  = (idx & 63) * 4;

        // feat rows are contiguous: async global -> LDS copy (ASYNCcnt)
        unsigned lds_off = (unsigned)(size_t)(&fs[row * XS_STRIDE + c4]);
        unsigned long long gaddr =
            (unsigned long long)(size_t)(feat + (size_t)(m_base + row) * N_IN + c4);
        asm volatile("global_load_async_to_lds_b128 %0, %1, off"
                     : : "v"(lds_off), "v"(gaddr) : "memory");

        // gathered x rows: synchronous float4 path
        int src = sampled[m_base + row];
        float4 xv = *(const float4*)(x + (size_t)src * N_IN + c4);
        *(float4*)&xs[row * XS_STRIDE + c4] = xv;
    }
    asm volatile("s_wait_asynccnt 0x0" ::: "memory");
    __syncthreads();

    // ---- Stage B: per-wave 16x16 WMMA tiles ------------------------------
    const int lane = t & 31;
    const int w    = t >> 5;          // wave 0..7 -> output cols 16w..16w+15
    const int half = lane >> 4;       // lanes 16-31 carry the K+2 pair
    const int l16  = lane & 15;
    const int nb   = w * 16;

    // self half: x[sampled] @ W_b + b_b  -> os cols [0,128)
    {
        const float bias = b_b[nb + l16];
        v8f acc;
        #pragma unroll
        for (int j = 0; j < 8; ++j) acc[j] = bias;
        const float* arow = &xs[l16 * XS_STRIDE];
        const float* bcol = WT_b + (size_t)(nb + l16) * N_IN;
        #pragma unroll 8
        for (int k0 = 0; k0 < N_IN; k0 += 4) {
            int kk = k0 + 2 * half;
            v2f a = *(const v2f*)(arow + kk);
            v2f b = *(const v2f*)(bcol + kk);
            acc = __builtin_amdgcn_wmma_f32_16x16x4_f32(
                false, a, false, b, (short)0, acc, false, false);
        }
        #pragma unroll
        for (int j = 0; j < 8; ++j)
            os[(j + 8 * half) * 256 + nb + l16] = acc[j];
    }

    // neigh half: feat @ W_w + b_w -> os cols [128,256)
    {
        const float bias = b_w[nb + l16];
        v8f acc;
        #pragma unroll
        for (int j = 0; j < 8; ++j) acc[j] = bias;
        const float* arow = &fs[l16 * XS_STRIDE];
        const float* bcol = WT_w + (size_t)(nb + l16) * N_IN;
        #pragma unroll 8
        for (int k0 = 0; k0 < N_IN; k0 += 4) {
            int kk = k0 + 2 * half;
            v2f a = *(const v2f*)(arow + kk);
            v2f b = *(const v2f*)(bcol + kk);
            acc = __builtin_amdgcn_wmma_f32_16x16x4_f32(
                false, a, false, b, (short)0, acc, false, false);
        }
        #pragma unroll
        for (int j = 0; j < 8; ++j)
            os[(j + 8 * half) * 256 + 128 + nb + l16] = acc[j];
    }
    __syncthreads();

    // ---- Stage C: ELU + row layernorm + affine, 2 rows per wave ----------
    #pragma unroll
    for (int rr = 0; rr < 2; ++rr) {
        const int r = w * 2 + rr;
        float v[8];
        float s1 = 0.f, s2 = 0.f;
        #pragma unroll
        for (int i = 0; i < 8; ++i) {
            float tv = os[r * 256 + lane + 32 * i];
            tv = (tv > 0.f) ? tv : (expf(tv) - 1.0f);   // ELU
            v[i] = tv;
            s1 += tv;
            s2 += tv * tv;
        }
        #pragma unroll
        for (int off = 16; off > 0; off >>= 1) {
            s1 += __shfl_xor(s1, off, 32);
            s2 += __shfl_xor(s2, off, 32);
        }
        const float mean = s1 * (1.0f / 256.0f);
        const float var  = s2 * (1.0f / 256.0f) - mean * mean + 1e-9f;
        const float inv  = rsqrtf(var);
        const size_t ob  = (size_t)(m_base + r) * 256;
        #pragma unroll
        for (int i = 0; i < 8; ++i) {
            int col = lane + 32 * i;
            out[ob + col] = (v[i] - mean) * scale[col] * inv + offset[col];
        }
    }
}

// ---------------------------------------------------------------------------
extern "C" void kernel_launch(void* const* d_in, const int* in_sizes, int n_in,
                              void* d_out, int out_size, void* d_ws, size_t ws_size,
                              hipStream_t stream) {
    const float* x       = (const float*)d_in[0];
    const int*   erows   = (const int*)  d_in[1];
    const int*   ecols   = (const int*)  d_in[2];
    const float* evals   = (const float*)d_in[3];
    const int*   sampled = (const int*)  d_in[4];
    const float* W_w     = (const float*)d_in[5];
    const float* b_w     = (const float*)d_in[6];
    const float* W_b     = (const float*)d_in[7];
    const float* b_b     = (const float*)d_in[8];
    const float* scale   = (const float*)d_in[9];
    const float* offset  = (const float*)d_in[10];
    float*       out     = (float*)d_out;

    // workspace layout: feat (51.2 MB) | WT_b (128 KB) | WT_w (128 KB)
    float* feat = (float*)d_ws;
    float* WT_b = feat + (size_t)N_DST * N_IN;
    float* WT_w = WT_b + (size_t)N_IN * N_OUT;

    // zero feat: 50000*256/4 = 3.2M float4 -> 12500 blocks exact
    sage_zero_feat<<<dim3(12500), dim3(256), 0, stream>>>((float4*)feat);

    // pack weights column-major for float2 B-fragment loads
    sage_pack_w<<<dim3(256), dim3(256), 0, stream>>>(W_b, W_w, WT_b, WT_w);

    // SpMM scatter: 800000 edges * 64 threads / 256 = 200000 blocks exact
    sage_spmm<<<dim3(200000), dim3(256), 0, stream>>>(x, erows, ecols, evals, feat);

    // fused dual-GEMM + ELU + layernorm: 50000/16 = 3125 blocks exact
    sage_fused<<<dim3(3125), dim3(256), 0, stream>>>(
        x, sampled, feat, WT_w, b_w, WT_b, b_b, scale, offset, out);
}